// DataPreprocessor_44770739093809
// MI455X (gfx1250) — compile-verified
//
#include <hip/hip_runtime.h>
#include <stdint.h>

namespace {

constexpr int B      = 8;
constexpr int N      = 120000;
constexpr int M      = 131072;   // next pow2 >= N (2^17)
constexpr int LOGM   = 17;
constexpr int FEAT   = 64;
constexpr int VOXW   = 68;       // 3 shifted + 1 intensity + 64 feats
constexpr uint32_t CAPY = 1602;
constexpr uint32_t CAPZ = 42;
constexpr uint32_t SENT = 94870441u;   // (1408+2)*1602*42 + 1

// ---------------- fill outputs: vox=0, coors=-1, valid=0 ----------------
__global__ void k_fill(float* __restrict__ out) {
  size_t i = (size_t)blockIdx.x * blockDim.x + threadIdx.x;
  constexpr size_t VEND = (size_t)B * N * VOXW;            // end of vox
  constexpr size_t CEND = (size_t)B * N * (VOXW + 4);      // end of coors
  constexpr size_t TOT  = (size_t)B * N * (VOXW + 5);      // + valid
  if (i >= TOT) return;
  out[i] = (i >= VEND && i < CEND) ? -1.0f : 0.0f;
}

// ---------------- key build: key = (hash << 32) | index ----------------
__global__ void k_keys(const float* __restrict__ points, uint64_t* __restrict__ keys) {
  size_t t = (size_t)blockIdx.x * blockDim.x + threadIdx.x;   // < B*M exactly
  int b = (int)(t >> LOGM);
  int i = (int)(t & (M - 1));
  uint64_t key = ~0ull;                                       // padding sorts last
  if (i < N) {
    float x = points[((size_t)b * 4 + 0) * N + i];
    float y = points[((size_t)b * 4 + 1) * N + i];
    float z = points[((size_t)b * 4 + 2) * N + i];
    bool inr = (x > 0.0f) && (x < 70.4f) &&
               (y > -40.0f) && (y < 40.0f) &&
               (z > -3.0f) && (z < 1.0f);
    uint32_t h = SENT;
    if (inr) {
      float sx = x - 0.0f, sy = y + 40.0f, sz = z + 3.0f;     // xyz - low
      int cx = (int)rintf(sx / 0.05f);                        // round-half-even
      int cy = (int)rintf(sy / 0.05f);
      int cz = (int)rintf(sz / 0.1f);
      h = ((uint32_t)cx * CAPY + (uint32_t)cy) * CAPZ + (uint32_t)cz;
    }
    key = ((uint64_t)h << 32) | (uint32_t)i;                  // tie-break = stability
  }
  keys[t] = key;
}

// ---------------- bitonic sort, global step (j >= 512) ----------------
__global__ void k_bitonic_g(uint64_t* __restrict__ keys, unsigned j, unsigned k) {
  size_t t = (size_t)blockIdx.x * blockDim.x + threadIdx.x;   // < B*M
  size_t b = t >> LOGM;
  unsigned i = (unsigned)(t & (M - 1));
  unsigned p = i ^ j;
  if (p <= i) return;
  uint64_t* kb = keys + (b << LOGM);
  uint64_t a = kb[i], c = kb[p];
  bool asc = ((i & k) == 0);
  if ((a > c) == asc) { kb[i] = c; kb[p] = a; }
}

// ---------------- bitonic sort, LDS-fused steps (j <= 256) ----------------
__global__ __launch_bounds__(256) void k_bitonic_l(uint64_t* __restrict__ keys,
                                                   unsigned k, unsigned jstart) {
  __shared__ uint64_t sh[512];
  constexpr int TILES = M / 512;
  int b    = blockIdx.x / TILES;
  int tile = blockIdx.x % TILES;
  unsigned baseI = (unsigned)tile * 512u;
  uint64_t* kb = keys + ((size_t)b << LOGM) + baseI;
  unsigned t = threadIdx.x;
  sh[t]       = kb[t];
  sh[t + 256] = kb[t + 256];
  __syncthreads();
  for (unsigned j = jstart; j >= 1; j >>= 1) {
    unsigned l = ((t & ~(j - 1)) << 1) | (t & (j - 1));       // partner pair (l, l|j)
    unsigned p = l | j;
    bool asc = (((baseI + l) & k) == 0);
    uint64_t a = sh[l], c = sh[p];
    if ((a > c) == asc) { sh[l] = c; sh[p] = a; }
    __syncthreads();
  }
  kb[t]       = sh[t];
  kb[t + 256] = sh[t + 256];
}

// ------- per-batch scan: mark first-of-run (h<SENT) and compute dests -------
__global__ __launch_bounds__(1024) void k_scan(const uint64_t* __restrict__ keys,
                                               uint32_t* __restrict__ pos) {
  __shared__ uint32_t sh[1024];
  constexpr int CHUNK = M / 1024;                              // 128
  int b = blockIdx.x, t = threadIdx.x;
  const uint64_t* kb = keys + ((size_t)b << LOGM);
  uint32_t* pb = pos + ((size_t)b << LOGM);
  int base = t * CHUNK;
  uint32_t prevh = (t == 0) ? 0xFFFFFFFFu : (uint32_t)(kb[base - 1] >> 32);
  uint32_t cnt = 0, ph = prevh;
  for (int e = 0; e < CHUNK; ++e) {
    uint32_t h = (uint32_t)(kb[base + e] >> 32);
    bool v = (h < SENT) && ((base + e == 0) || (h != ph));
    cnt += v; ph = h;
  }
  sh[t] = cnt; __syncthreads();
  for (int off = 1; off < 1024; off <<= 1) {                   // Kogge-Stone inclusive
    uint32_t v = (t >= off) ? sh[t - off] : 0u;
    __syncthreads();
    sh[t] += v;
    __syncthreads();
  }
  uint32_t run = sh[t] - cnt;                                  // exclusive start
  ph = prevh;
  for (int e = 0; e < CHUNK; ++e) {
    uint32_t h = (uint32_t)(kb[base + e] >> 32);
    bool v = (h < SENT) && ((base + e == 0) || (h != ph));
    pb[base + e] = v ? run : 0xFFFFFFFFu;
    run += v; ph = h;
  }
}

// ------- scatter: 64 lanes per voxel; async global->LDS feature gather -------
__global__ __launch_bounds__(256) void k_scatter(const uint64_t* __restrict__ keys,
                                                 const uint32_t* __restrict__ pos,
                                                 const float* __restrict__ points,
                                                 const float* __restrict__ feats,
                                                 float* __restrict__ out) {
  __shared__ float staging[256];
  int grp  = threadIdx.x >> 6;                 // 4 groups of 64 (= 2 full waves each)
  int lane = threadIdx.x & 63;
  size_t slot = (size_t)blockIdx.x * 4 + grp;
  uint32_t p = pos[slot];
  if (p == 0xFFFFFFFFu) return;                // wave-uniform branch (group = 2 waves)
  int b = (int)(slot >> LOGM);
  uint32_t idx = (uint32_t)keys[slot];         // original point index
  size_t row = (size_t)b * N + p;              // compacted destination row
  float* vrow = out + row * VOXW;

  // Issue the per-lane feature gather asynchronously into LDS, then overlap
  // the scalar metadata work before waiting on ASYNCcnt.
  const float* g = feats + ((size_t)b * FEAT + lane) * N + idx;
  uint32_t lds_off = (uint32_t)(uintptr_t)&staging[threadIdx.x];
  asm volatile("global_load_async_to_lds_b32 %0, %1, off"
               :: "v"(lds_off), "v"(g) : "memory");

  if (lane == 0) {
    float x = points[((size_t)b * 4 + 0) * N + idx];
    float y = points[((size_t)b * 4 + 1) * N + idx];
    float z = points[((size_t)b * 4 + 2) * N + idx];
    float w = points[((size_t)b * 4 + 3) * N + idx];
    float sx = x - 0.0f, sy = y + 40.0f, sz = z + 3.0f;
    int cx = (int)rintf(sx / 0.05f);
    int cy = (int)rintf(sy / 0.05f);
    int cz = (int)rintf(sz / 0.1f);
    vrow[0] = sx; vrow[1] = sy; vrow[2] = sz; vrow[3] = w;
    float* crow = out + (size_t)B * N * VOXW + row * 4;
    crow[0] = (float)b; crow[1] = (float)cz; crow[2] = (float)cy; crow[3] = (float)cx;
    out[(size_t)B * N * (VOXW + 4) + row] = 1.0f;
  }

#if __has_builtin(__builtin_amdgcn_s_wait_asynccnt)
  __builtin_amdgcn_s_wait_asynccnt(0);
#else
  asm volatile("s_wait_asynccnt 0" ::: "memory");
#endif
  vrow[4 + lane] = staging[threadIdx.x];       // each lane reads its own slot
}

} // anonymous namespace

extern "C" void kernel_launch(void* const* d_in, const int* in_sizes, int n_in,
                              void* d_out, int out_size, void* d_ws, size_t ws_size,
                              hipStream_t stream) {
  const float* points = (const float*)d_in[0];   // (B, 4, N)
  const float* feats  = (const float*)d_in[1];   // (B, 64, N)
  float* out = (float*)d_out;                    // [vox | coors | valid] as float32

  uint64_t* keys = (uint64_t*)d_ws;                                        // B*M*8  = 8 MB
  uint32_t* pos  = (uint32_t*)((char*)d_ws + (size_t)B * M * sizeof(uint64_t)); // B*M*4 = 4 MB

  constexpr size_t TOT = (size_t)B * N * (VOXW + 5);
  k_fill<<<(unsigned)((TOT + 255) / 256), 256, 0, stream>>>(out);
  k_keys<<<(B * M) / 256, 256, 0, stream>>>(points, keys);

  // Bitonic merge network over M=2^17 per batch; j>=512 global, j<=256 fused in LDS.
  for (unsigned k = 2; k <= (unsigned)M; k <<= 1) {
    unsigned j = k >> 1;
    for (; j >= 512; j >>= 1)
      k_bitonic_g<<<(B * M) / 256, 256, 0, stream>>>(keys, j, k);
    k_bitonic_l<<<B * (M / 512), 256, 0, stream>>>(keys, k, j);
  }

  k_scan<<<B, 1024, 0, stream>>>(keys, pos);
  k_scatter<<<(B * M) / 4, 256, 0, stream>>>(keys, pos, points, feats, out);
}